// LinearProbe_28862180229503
// MI455X (gfx1250) — compile-verified
//
#include <hip/hip_runtime.h>
#include <hip/hip_bf16.h>
#include <math.h>
#include <stdint.h>

// ---------------------------------------------------------------------------
// LOO ridge probe, restructured:
//   S = X^T X, T = X^T Y (once, WMMA f32 GEMM)
//   per i:  G_i = D_i (S - x_i x_i^T - 511 mu_i mu_i^T) D_i + eps I   (rank-2)
//           B_i = D_i (T - x_i y_i^T - 511 mu_i nu_i^T) E_i
//           solve G_i beta = B_i via blocked Cholesky (WMMA f32 16x16x4)
//           out_i = sigma_y .* (xnorm_i . beta) + mu_y
// Chunked so 128 Gram matrices (128 MB) fit in the 192 MB L2.
// gfx1250 paths: v_wmma_f32_16x16x4_f32 everywhere matrix math happens,
// global_load_async_to_lds_b128 staging overlapped with serial diag work.
// ---------------------------------------------------------------------------

typedef __attribute__((ext_vector_type(2))) float v2f;
typedef __attribute__((ext_vector_type(8))) float v8f;

#define N 512
#define D 512
#define O 32
#define CHUNK 128
#define EPSILON 1.0f

// Async DMA: copy 16 bytes global -> LDS, tracked by ASYNCcnt.
__device__ __forceinline__ void async_copy_b128(void* lds_dst, const float* gsrc) {
  const uint32_t lds_addr = (uint32_t)(uintptr_t)lds_dst;   // low 32b of flat = LDS offset
  const uint64_t gaddr = (uint64_t)(uintptr_t)gsrc;
  asm volatile("global_load_async_to_lds_b128 %0, %1, off"
               :: "v"(lds_addr), "v"(gaddr) : "memory");
}
__device__ __forceinline__ void wait_async0() {
  asm volatile("s_wait_asynccnt 0x0" ::: "memory");
}

// D(16x16) += A(16x16) x B(16x16) as 4 chained V_WMMA_F32_16X16X4_F32.
// A 16x4 layout: lanes 0-15 -> M=lane, VGPR0=K0,VGPR1=K1; lanes 16-31 -> K2,K3.
// B 4x16 layout: row-striped across lanes per VGPR (mirror of A).
template <typename FA, typename FB>
__device__ __forceinline__ v8f mma_16x16x16(FA la, FB lb, int lane, v8f acc) {
  const int hi = (lane >> 4) & 1;
  const int lo = lane & 15;
#pragma unroll
  for (int s = 0; s < 4; ++s) {
    const int k0 = s * 4 + 2 * hi;
    v2f a, b;
    a.x = la(lo, k0);
    a.y = la(lo, k0 + 1);
    b.x = lb(k0, lo);
    b.y = lb(k0 + 1, lo);
    acc = __builtin_amdgcn_wmma_f32_16x16x4_f32(false, a, false, b, (short)0,
                                                acc, false, false);
  }
  return acc;
}

// C/D 16x16 layout: VGPR r -> M=r (lanes 0-15), M=r+8 (lanes 16-31); N=lane&15.
template <typename FS>
__device__ __forceinline__ void store_16x16(v8f c, FS st, int lane) {
  const int hi = (lane >> 4) & 1;
  const int lo = lane & 15;
#pragma unroll
  for (int r = 0; r < 8; ++r) st(r + 8 * hi, lo, c[r]);
}

// ---------------- column sums / sums of squares --------------------------
__global__ __launch_bounds__(256) void k_colstats(
    const float* __restrict__ X, const float* __restrict__ Y,
    float* __restrict__ sumX, float* __restrict__ sumX2,
    float* __restrict__ sumY, float* __restrict__ sumY2) {
  __shared__ float s1[256], s2[256];
  const int t = threadIdx.x;
  const int b = blockIdx.x;
  float v0, v1;
  if (b < D) {
    v0 = X[(size_t)t * D + b];
    v1 = X[(size_t)(t + 256) * D + b];
  } else {
    const int o = b - D;
    v0 = Y[(size_t)t * O + o];
    v1 = Y[(size_t)(t + 256) * O + o];
  }
  s1[t] = v0 + v1;
  s2[t] = v0 * v0 + v1 * v1;
  __syncthreads();
  for (int st = 128; st > 0; st >>= 1) {
    if (t < st) { s1[t] += s1[t + st]; s2[t] += s2[t + st]; }
    __syncthreads();
  }
  if (t == 0) {
    if (b < D) { sumX[b] = s1[0]; sumX2[b] = s2[0]; }
    else       { sumY[b - D] = s1[0]; sumY2[b - D] = s2[0]; }
  }
}

// ---------------- per-sample LOO stats -----------------------------------
__global__ __launch_bounds__(256) void k_perstats(
    const float* __restrict__ X, const float* __restrict__ Y,
    const float* __restrict__ sumX, const float* __restrict__ sumX2,
    const float* __restrict__ sumY, const float* __restrict__ sumY2,
    float* __restrict__ muX, float* __restrict__ rsX, float* __restrict__ XN,
    float* __restrict__ muY, float* __restrict__ rsY, float* __restrict__ sigY) {
  const int i = blockIdx.x, t = threadIdx.x;
  for (int d = t; d < D; d += 256) {
    const float x = X[(size_t)i * D + d];
    const float mu = (sumX[d] - x) * (1.f / 511.f);
    const float var = (sumX2[d] - x * x - 511.f * mu * mu) * (1.f / 510.f);
    const float rs = rsqrtf(var);
    muX[(size_t)i * D + d] = mu;
    rsX[(size_t)i * D + d] = rs;
    XN[(size_t)i * D + d] = (x - mu) * rs;
  }
  if (t < O) {
    const float y = Y[(size_t)i * O + t];
    const float mu = (sumY[t] - y) * (1.f / 511.f);
    const float var = (sumY2[t] - y * y - 511.f * mu * mu) * (1.f / 510.f);
    const float sg = sqrtf(var);
    muY[(size_t)i * O + t] = mu;
    rsY[(size_t)i * O + t] = 1.f / sg;
    sigY[(size_t)i * O + t] = sg;
  }
}

// ---------------- S = X^T X (one wave per 16x16 tile) --------------------
__global__ __launch_bounds__(32) void k_gemm_S(const float* __restrict__ X,
                                               float* __restrict__ S) {
  const int di = blockIdx.x * 16, dj = blockIdx.y * 16;
  const int lane = threadIdx.x;
  v8f acc = {};
  for (int k0 = 0; k0 < N; k0 += 16) {
    acc = mma_16x16x16(
        [&](int m, int kk) { return X[(size_t)(k0 + kk) * D + di + m]; },
        [&](int kk, int n) { return X[(size_t)(k0 + kk) * D + dj + n]; },
        lane, acc);
  }
  store_16x16(acc,
              [&](int m, int n, float v) { S[(size_t)(di + m) * D + dj + n] = v; },
              lane);
}

// ---------------- T = X^T Y ----------------------------------------------
__global__ __launch_bounds__(32) void k_gemm_T(const float* __restrict__ X,
                                               const float* __restrict__ Y,
                                               float* __restrict__ T) {
  const int di = blockIdx.x * 16, dj = blockIdx.y * 16;
  const int lane = threadIdx.x;
  v8f acc = {};
  for (int k0 = 0; k0 < N; k0 += 16) {
    acc = mma_16x16x16(
        [&](int m, int kk) { return X[(size_t)(k0 + kk) * D + di + m]; },
        [&](int kk, int n) { return Y[(size_t)(k0 + kk) * O + dj + n]; },
        lane, acc);
  }
  store_16x16(acc,
              [&](int m, int n, float v) { T[(size_t)(di + m) * O + dj + n] = v; },
              lane);
}

// ---------------- build G_i (rank-2 corrected, scaled, ridged) -----------
// S rows are staged to LDS with async-DMA, overlapped with the stat loads.
__global__ __launch_bounds__(256) void k_buildG(
    const float* __restrict__ S, const float* __restrict__ X,
    const float* __restrict__ muX, const float* __restrict__ rsX,
    float* __restrict__ G, int cbase) {
  __shared__ float sS[16 * D];          // 32 KB staged S row-slab
  __shared__ float xe[D], mue[D], rse[D];
  const int i = cbase + blockIdx.x;
  float* Gi = G + (size_t)blockIdx.x * D * D;
  const int t = threadIdx.x;
  const int r0 = blockIdx.y * 16;

  // kick DMA: 16 rows x 512 floats = 2048 16B-chunks, 8 per thread
  for (int idx = t; idx < 16 * (D / 4); idx += 256) {
    const int r = idx >> 7;              // D/4 = 128 chunks per row
    const int c4 = (idx & 127) * 4;
    async_copy_b128(&sS[r * D + c4], S + (size_t)(r0 + r) * D + c4);
  }
  // overlap: per-sample stat vectors via normal loads
  for (int e = t; e < D; e += 256) {
    xe[e] = X[(size_t)i * D + e];
    mue[e] = muX[(size_t)i * D + e];
    rse[e] = rsX[(size_t)i * D + e];
  }
  wait_async0();
  __syncthreads();

  const int dr = t >> 4;
  const int d = r0 + dr;
  const int e0 = t & 15;
  const float xd = xe[d], mud = mue[d], rsd = rse[d];
  const float* Srow = &sS[dr * D];
  float* Grow = Gi + (size_t)d * D;
#pragma unroll 4
  for (int j = 0; j < 32; ++j) {
    const int e = e0 + j * 16;
    float g = (Srow[e] - xd * xe[e] - 511.f * mud * mue[e]) * (rsd * rse[e]);
    if (e == d) g += EPSILON;
    Grow[e] = g;
  }
}

// ---------------- batched blocked Cholesky + solves + output -------------
__global__ __launch_bounds__(256) void k_chol_solve(
    float* __restrict__ G, const float* __restrict__ T,
    const float* __restrict__ X, const float* __restrict__ Y,
    const float* __restrict__ muX, const float* __restrict__ rsX,
    const float* __restrict__ XN, const float* __restrict__ muY,
    const float* __restrict__ rsY, const float* __restrict__ sigY,
    float* __restrict__ out, int cbase) {
  __shared__ float sInvL[32 * 256];   // inv(L_kk) for every diagonal block
  __shared__ float sPanel[31 * 256];  // current panel L blocks
  __shared__ float sA[31 * 256];      // async-staged A column for panel TRSM
  __shared__ float sB[D * O];         // RHS -> z -> beta (in place)
  __shared__ float sDiag[256];
  __shared__ float sRed[256];

  const int i = cbase + blockIdx.x;
  float* Gi = G + (size_t)blockIdx.x * D * D;
  const int t = threadIdx.x;
  const int wave = t >> 5;
  const int lane = t & 31;

  // RHS: B_i[d][o] = (T - x_d y_o - 511 mu_d nu_o) * rs_d * rsy_o
  for (int idx = t; idx < D * O; idx += 256) {
    const int d = idx >> 5, o = idx & 31;
    sB[idx] = (T[idx] - X[(size_t)i * D + d] * Y[(size_t)i * O + o] -
               511.f * muX[(size_t)i * D + d] * muY[(size_t)i * O + o]) *
              (rsX[(size_t)i * D + d] * rsY[(size_t)i * O + o]);
  }
  __syncthreads();

  // ---- blocked Cholesky, NB = 16, 32 panel steps ----
  for (int k = 0; k < 32; ++k) {
    const int kb = k * 16;
    sDiag[t] = Gi[(size_t)(kb + (t >> 4)) * D + kb + (t & 15)];

    // kick async DMA of this step's panel A column (rows below diag block);
    // it fills sA while lane 0 runs the serial 16x16 factor+inverse below.
    const int nch = (31 - k) * 64;      // rows * 4 chunks of 16B
    for (int idx = t; idx < nch; idx += 256) {
      const int row = idx >> 2;
      const int c4 = (idx & 3) * 4;
      async_copy_b128(&sA[row * 16 + c4],
                      Gi + (size_t)(kb + 16 + row) * D + kb + c4);
    }
    __syncthreads();

    if (t == 0) {
      // 16x16 lower Cholesky (overlapped with the ASYNC DMA above)
      for (int j = 0; j < 16; ++j) {
        const float djj = sqrtf(sDiag[j * 16 + j]);
        sDiag[j * 16 + j] = djj;
        const float inv = 1.f / djj;
        for (int r = j + 1; r < 16; ++r) sDiag[r * 16 + j] *= inv;
        for (int c = j + 1; c < 16; ++c) {
          const float ljc = sDiag[c * 16 + j];
          for (int r = c; r < 16; ++r) sDiag[r * 16 + c] -= sDiag[r * 16 + j] * ljc;
        }
      }
      // W = inv(L_kk), lower triangular
      float* W = &sInvL[k * 256];
      for (int c = 0; c < 16; ++c) {
        for (int r = 0; r < c; ++r) W[r * 16 + c] = 0.f;
        W[c * 16 + c] = 1.f / sDiag[c * 16 + c];
        for (int r = c + 1; r < 16; ++r) {
          float s = 0.f;
          for (int j2 = c; j2 < r; ++j2) s += sDiag[r * 16 + j2] * W[j2 * 16 + c];
          W[r * 16 + c] = -s / sDiag[r * 16 + r];
        }
      }
    }
    wait_async0();
    __syncthreads();

    // panel TRSM as GEMM: L[bi][k] = A[bi][k] * W^T  (A from staged LDS)
    const float* W = &sInvL[k * 256];
    for (int bi = k + 1 + wave; bi < 32; bi += 8) {
      const float* Ablk = &sA[(bi - k - 1) * 256];
      v8f acc = {};
      acc = mma_16x16x16(
          [&](int m, int kk) { return Ablk[m * 16 + kk]; },
          [&](int kk, int n) { return W[n * 16 + kk]; }, lane, acc);
      float* Pblk = &sPanel[(bi - k - 1) * 256];
      float* Lblk = Gi + (size_t)(bi * 16) * D + kb;
      store_16x16(acc,
                  [&](int m, int n, float v) {
                    Pblk[m * 16 + n] = v;
                    Lblk[(size_t)m * D + n] = v;
                  },
                  lane);
    }
    __syncthreads();

    // trailing SYRK: G[bi][bj] -= P_bi * P_bj^T   (WMMA)
    const int tr = 31 - k;
    const int ntiles = tr * (tr + 1) / 2;
    for (int idx = wave; idx < ntiles; idx += 8) {
      int a = (int)((sqrtf(8.f * (float)idx + 1.f) - 1.f) * 0.5f);
      while ((a + 1) * (a + 2) / 2 <= idx) ++a;
      while (a * (a + 1) / 2 > idx) --a;
      const int b = idx - a * (a + 1) / 2;
      const int bi = k + 1 + a, bj = k + 1 + b;
      const float* Pa = &sPanel[a * 256];
      const float* Pb = &sPanel[b * 256];
      v8f acc = {};
      acc = mma_16x16x16(
          [&](int m, int kk) { return Pa[m * 16 + kk]; },
          [&](int kk, int n) { return Pb[n * 16 + kk]; }, lane, acc);
      float* Cblk = Gi + (size_t)(bi * 16) * D + bj * 16;
      store_16x16(acc,
                  [&](int m, int n, float v) { Cblk[(size_t)m * D + n] -= v; },
                  lane);
    }
    // make the trailing stores visible to next step's loads / async DMA
    __threadfence_block();
    __syncthreads();
  }
  __threadfence_block();
  __syncthreads();

  // ---- forward solve L z = B ----
  for (int k = 0; k < 32; ++k) {
    const int kb = k * 16;
    const float* W = &sInvL[k * 256];
    if (wave < 2) {
      const int h = wave;
      v8f acc = {};
      acc = mma_16x16x16(
          [&](int m, int kk) { return W[m * 16 + kk]; },
          [&](int kk, int n) { return sB[(kb + kk) * O + h * 16 + n]; }, lane, acc);
      store_16x16(acc,
                  [&](int m, int n, float v) { sB[(kb + m) * O + h * 16 + n] = v; },
                  lane);
    }
    __syncthreads();
    const int nt = 2 * (31 - k);
    for (int idx = wave; idx < nt; idx += 8) {
      const int bi = k + 1 + (idx >> 1);
      const int h = idx & 1;
      const float* Lblk = Gi + (size_t)(bi * 16) * D + kb;
      v8f acc = {};
      acc = mma_16x16x16(
          [&](int m, int kk) { return Lblk[(size_t)m * D + kk]; },
          [&](int kk, int n) { return sB[(kb + kk) * O + h * 16 + n]; }, lane, acc);
      store_16x16(acc,
                  [&](int m, int n, float v) { sB[(bi * 16 + m) * O + h * 16 + n] -= v; },
                  lane);
    }
    __syncthreads();
  }

  // ---- backward solve L^T beta = z ----
  for (int k = 31; k >= 0; --k) {
    const int kb = k * 16;
    const float* W = &sInvL[k * 256];
    if (wave < 2) {
      const int h = wave;
      v8f acc = {};
      acc = mma_16x16x16(
          [&](int m, int kk) { return W[kk * 16 + m]; },  // W^T
          [&](int kk, int n) { return sB[(kb + kk) * O + h * 16 + n]; }, lane, acc);
      store_16x16(acc,
                  [&](int m, int n, float v) { sB[(kb + m) * O + h * 16 + n] = v; },
                  lane);
    }
    __syncthreads();
    const int nt = 2 * k;
    for (int idx = wave; idx < nt; idx += 8) {
      const int bi = idx >> 1;
      const int h = idx & 1;
      const float* Lblk = Gi + (size_t)kb * D + bi * 16;  // L[k][bi], used transposed
      v8f acc = {};
      acc = mma_16x16x16(
          [&](int m, int kk) { return Lblk[(size_t)kk * D + m]; },
          [&](int kk, int n) { return sB[(kb + kk) * O + h * 16 + n]; }, lane, acc);
      store_16x16(acc,
                  [&](int m, int n, float v) { sB[(bi * 16 + m) * O + h * 16 + n] -= v; },
                  lane);
    }
    __syncthreads();
  }

  // ---- output: out = sigma_y .* (xnorm . beta) + mu_y ----
  {
    const int o = t & 31;
    const int sl = t >> 5;
    float p = 0.f;
    for (int d = sl * 64; d < sl * 64 + 64; ++d)
      p += XN[(size_t)i * D + d] * sB[d * O + o];
    sRed[t] = p;
    __syncthreads();
    if (t < O) {
      float s = 0.f;
      for (int w2 = 0; w2 < 8; ++w2) s += sRed[w2 * 32 + t];
      out[(size_t)i * O + t] = sigY[(size_t)i * O + t] * s + muY[(size_t)i * O + t];
    }
  }
}

// ---------------------------------------------------------------------------
extern "C" void kernel_launch(void* const* d_in, const int* in_sizes, int n_in,
                              void* d_out, int out_size, void* d_ws, size_t ws_size,
                              hipStream_t stream) {
  (void)in_sizes; (void)n_in; (void)out_size; (void)ws_size;
  const float* X = (const float*)d_in[0];  // (512,512)
  const float* Y = (const float*)d_in[1];  // (512,32)
  float* out = (float*)d_out;              // (512,32)

  float* w = (float*)d_ws;
  size_t off = 0;
  float* S     = w + off; off += (size_t)D * D;
  float* T     = w + off; off += (size_t)D * O;
  float* sumX  = w + off; off += D;
  float* sumX2 = w + off; off += D;
  float* sumY  = w + off; off += O;
  float* sumY2 = w + off; off += O;
  float* muX   = w + off; off += (size_t)N * D;
  float* rsX   = w + off; off += (size_t)N * D;
  float* XNrm  = w + off; off += (size_t)N * D;
  float* muY   = w + off; off += (size_t)N * O;
  float* rsY   = w + off; off += (size_t)N * O;
  float* sigY  = w + off; off += (size_t)N * O;
  float* G     = w + off; off += (size_t)CHUNK * D * D;  // 128 MB: fits L2

  k_colstats<<<D + O, 256, 0, stream>>>(X, Y, sumX, sumX2, sumY, sumY2);
  k_perstats<<<N, 256, 0, stream>>>(X, Y, sumX, sumX2, sumY, sumY2,
                                    muX, rsX, XNrm, muY, rsY, sigY);
  k_gemm_S<<<dim3(D / 16, D / 16), 32, 0, stream>>>(X, S);
  k_gemm_T<<<dim3(D / 16, O / 16), 32, 0, stream>>>(X, Y, T);

  for (int c = 0; c < N / CHUNK; ++c) {
    k_buildG<<<dim3(CHUNK, D / 16), 256, 0, stream>>>(S, X, muX, rsX, G, c * CHUNK);
    k_chol_solve<<<CHUNK, 256, 0, stream>>>(G, T, X, Y, muX, rsX, XNrm,
                                            muY, rsY, sigY, out, c * CHUNK);
  }
}